// TransitionDown_88407606821407
// MI455X (gfx1250) — compile-verified
//
#include <hip/hip_runtime.h>

#define Bc   16
#define Pc   4096
#define Mc   1024
#define Kc   16
#define FIN  64
#define FOUT 128
#define Nc   (Bc * Pc)
#define BN_EPS 1e-5f
#define FMAXV 3.402823466e38f

typedef float v2f __attribute__((ext_vector_type(2)));
typedef float v8f __attribute__((ext_vector_type(8)));
typedef int   v4i __attribute__((ext_vector_type(4)));

// ---------------------------------------------------------------------------
// CDNA5 async global->LDS path (ASYNCcnt-tracked DMA), with sync fallback so
// the file compiles on toolchains where the builtin is absent/renamed.
// Builtin signature (from hipcc diagnostics): (int4 AS1*, int4 AS3*, imm, imm).
// ---------------------------------------------------------------------------
#if defined(__gfx1250__) &&                                         \
    __has_builtin(__builtin_amdgcn_global_load_async_to_lds_b128) && \
    __has_builtin(__builtin_amdgcn_s_wait_asynccnt)
#define USE_ASYNC_LDS 1
#else
#define USE_ASYNC_LDS 0
#endif

// Copy 8192 contiguous floats (32 KB) global -> LDS with 256 threads.
__device__ __forceinline__ void copy_chunk(const float* __restrict__ src,
                                           float* __restrict__ dst, int tid) {
#if USE_ASYNC_LDS
#pragma unroll
  for (int i = 0; i < 8; ++i) {
    int off = (tid + i * 256) * 4;
    __builtin_amdgcn_global_load_async_to_lds_b128(
        (__attribute__((address_space(1))) v4i*)(src + off),
        (__attribute__((address_space(3))) v4i*)(dst + off), 0, 0);
  }
#else
#pragma unroll
  for (int i = 0; i < 8; ++i) {
    int off = (tid + i * 256) * 4;
    *(float4*)(dst + off) = *(const float4*)(src + off);
  }
#endif
}

__device__ __forceinline__ void async_wait() {
#if USE_ASYNC_LDS
  __builtin_amdgcn_s_wait_asynccnt(0);
#endif
}

// ---------------------------------------------------------------------------
// 1) Farthest point sampling: one 1024-thread block per cloud.
//    4 points per thread in registers, two-level argmax (shfl_xor + LDS).
// ---------------------------------------------------------------------------
__global__ __launch_bounds__(1024)
void fps_kernel(const float* __restrict__ pos, int* __restrict__ fps_idx,
                float* __restrict__ out_pos, int* __restrict__ out_batch) {
  const int b    = blockIdx.x;
  const int tid  = threadIdx.x;
  const int lane = tid & 31;
  const int wave = tid >> 5;

  __shared__ float s_bd[32];
  __shared__ int   s_bi[32];
  __shared__ float s_last[3];
  __shared__ int   s_fps[Mc];

  float px[4], py[4], pz[4], dmin[4];
#pragma unroll
  for (int i = 0; i < 4; ++i) {
    int p = tid + i * 1024;
    const float* pp = pos + (size_t)(b * Pc + p) * 3;
    px[i] = pp[0]; py[i] = pp[1]; pz[i] = pp[2];
    dmin[i] = FMAXV;
  }
  if (tid == 0) {
    const float* pp = pos + (size_t)(b * Pc) * 3;
    s_last[0] = pp[0]; s_last[1] = pp[1]; s_last[2] = pp[2];
    s_fps[0] = 0;
    fps_idx[b * Mc] = 0;
  }
  __syncthreads();

  for (int m = 1; m < Mc; ++m) {
    const float lx = s_last[0], ly = s_last[1], lz = s_last[2];
    float bd = -1.0f; int bi = 0x7fffffff;
#pragma unroll
    for (int i = 0; i < 4; ++i) {
      float dx = px[i] - lx, dy = py[i] - ly, dz = pz[i] - lz;
      float d = dx * dx + dy * dy + dz * dz;
      dmin[i] = fminf(dmin[i], d);
      int p = tid + i * 1024;
      if (dmin[i] > bd || (dmin[i] == bd && p < bi)) { bd = dmin[i]; bi = p; }
    }
#pragma unroll
    for (int off = 16; off >= 1; off >>= 1) {
      float od = __shfl_xor(bd, off, 32);
      int   oi = __shfl_xor(bi, off, 32);
      if (od > bd || (od == bd && oi < bi)) { bd = od; bi = oi; }
    }
    if (lane == 0) { s_bd[wave] = bd; s_bi[wave] = bi; }
    __syncthreads();
    if (wave == 0) {
      bd = s_bd[lane]; bi = s_bi[lane];
#pragma unroll
      for (int off = 16; off >= 1; off >>= 1) {
        float od = __shfl_xor(bd, off, 32);
        int   oi = __shfl_xor(bi, off, 32);
        if (od > bd || (od == bd && oi < bi)) { bd = od; bi = oi; }
      }
      if (lane == 0) {
        s_fps[m] = bi;
        fps_idx[b * Mc + m] = bi;
        const float* pp = pos + (size_t)(b * Pc + bi) * 3;
        s_last[0] = pp[0]; s_last[1] = pp[1]; s_last[2] = pp[2];
      }
    }
    __syncthreads();
  }

  for (int m = tid; m < Mc; m += 1024) {
    int li = s_fps[m];
    const float* pp = pos + (size_t)(b * Pc + li) * 3;
    int row = b * Mc + m;
    out_pos[row * 3 + 0] = pp[0];
    out_pos[row * 3 + 1] = pp[1];
    out_pos[row * 3 + 2] = pp[2];
    out_batch[row] = b;
  }
}

// ---------------------------------------------------------------------------
// 2) h = feat @ W + b via V_WMMA_F32_16X16X4_F32, with fused BN statistics.
//    256 blocks x 256 threads (8 waves). W staged in LDS via async DMA.
// ---------------------------------------------------------------------------
__global__ __launch_bounds__(256)
void gemm_bn_kernel(const float* __restrict__ feat, const float* __restrict__ W,
                    const float* __restrict__ bias, float* __restrict__ h,
                    float* __restrict__ g_sum, float* __restrict__ g_sq) {
  __shared__ __align__(16) float sW[FIN * FOUT];
  __shared__ float sB[FOUT];
  __shared__ float s_sum[FOUT];
  __shared__ float s_sq[FOUT];

  const int tid = threadIdx.x;
  copy_chunk(W, sW, tid);  // 32 KB async -> LDS
  if (tid < FOUT) { sB[tid] = bias[tid]; s_sum[tid] = 0.0f; s_sq[tid] = 0.0f; }
  async_wait();
  __syncthreads();

  const int wave  = tid >> 5;
  const int lane  = tid & 31;
  const int mn    = lane & 15;          // A row / B col within tile
  const int koff  = (lane >> 4) * 2;    // K sub-offset per A/B f32 layout
  const int radd  = (lane >> 4) * 8;    // C/D row add for upper half-wave
  const int col0  = wave * 16;

  float lsum = 0.0f, lsq = 0.0f;

  for (int j = 0; j < 16; ++j) {
    const int row0 = (blockIdx.x * 16 + j) * 16;
    v8f acc = {};
    for (int k0 = 0; k0 < FIN; k0 += 4) {
      v2f a, bf;
      const float* ap = feat + (size_t)(row0 + mn) * FIN + k0 + koff;
      a.x  = ap[0];
      a.y  = ap[1];
      bf.x = sW[(k0 + koff) * FOUT + col0 + mn];
      bf.y = sW[(k0 + koff + 1) * FOUT + col0 + mn];
      acc = __builtin_amdgcn_wmma_f32_16x16x4_f32(false, a, false, bf,
                                                  (short)0, acc, false, false);
    }
    const float bb = sB[col0 + mn];
#pragma unroll
    for (int i = 0; i < 8; ++i) {
      float v = acc[i] + bb;
      int row = row0 + i + radd;
      h[(size_t)row * FOUT + col0 + mn] = v;
      lsum += v; lsq += v * v;
    }
  }
  atomicAdd(&s_sum[col0 + mn], lsum);
  atomicAdd(&s_sq[col0 + mn], lsq);
  __syncthreads();
  if (tid < FOUT) {
    atomicAdd(&g_sum[tid], s_sum[tid]);
    atomicAdd(&g_sq[tid],  s_sq[tid]);
  }
}

__global__ void init_bn_kernel(float* g_sum, float* g_sq) {
  int c = threadIdx.x;
  if (c < FOUT) { g_sum[c] = 0.0f; g_sq[c] = 0.0f; }
}

__global__ void bn_finalize_kernel(const float* __restrict__ g_sum,
                                   const float* __restrict__ g_sq,
                                   const float* __restrict__ gamma,
                                   const float* __restrict__ beta,
                                   float* __restrict__ scale,
                                   float* __restrict__ shift) {
  int c = threadIdx.x;
  if (c < FOUT) {
    float mean = g_sum[c] * (1.0f / (float)Nc);
    float var  = g_sq[c] * (1.0f / (float)Nc) - mean * mean;
    float sc   = gamma[c] * rsqrtf(var + BN_EPS);
    scale[c] = sc;
    shift[c] = beta[c] - mean * sc;
  }
}

__global__ __launch_bounds__(256)
void bn_apply_kernel(float* __restrict__ h, const float* __restrict__ scale,
                     const float* __restrict__ shift) {
  size_t gid = (size_t)blockIdx.x * 256 + threadIdx.x;  // one float4 each
  float4 v = ((const float4*)h)[gid];
  int c0 = (int)((gid * 4) & (FOUT - 1));
  v.x = fmaxf(v.x * scale[c0 + 0] + shift[c0 + 0], 0.0f);
  v.y = fmaxf(v.y * scale[c0 + 1] + shift[c0 + 1], 0.0f);
  v.z = fmaxf(v.z * scale[c0 + 2] + shift[c0 + 2], 0.0f);
  v.w = fmaxf(v.w * scale[c0 + 3] + shift[c0 + 3], 0.0f);
  ((float4*)h)[gid] = v;
}

// ---------------------------------------------------------------------------
// 2b) Precompute |x|^2 for every point (KNN rank term shared by 64 blocks).
// ---------------------------------------------------------------------------
__global__ __launch_bounds__(256)
void xnorm_kernel(const float* __restrict__ feat, float* __restrict__ xnorm) {
  int p = blockIdx.x * 256 + threadIdx.x;
  const float* fp = feat + (size_t)p * FIN;
  float s = 0.0f;
#pragma unroll
  for (int k = 0; k < FIN; k += 4) {
    float4 v = *(const float4*)(fp + k);
    s += v.x * v.x + v.y * v.y + v.z * v.z + v.w * v.w;
  }
  xnorm[p] = s;
}

// ---------------------------------------------------------------------------
// 3) KNN in feature space. Block = (cloud, 16-query tile). Double-buffered
//    128-point X chunks streamed into LDS via GLOBAL_LOAD_ASYNC_TO_LDS_B128,
//    overlapped with f32 WMMA distance tiles (d2 = |x|^2 - 2 q.x; |q|^2 is
//    rank-invariant) and insertion top-16 per query.
// ---------------------------------------------------------------------------
__global__ __launch_bounds__(256)
void knn_kernel(const float* __restrict__ feat, const float* __restrict__ xnorm,
                const int* __restrict__ fps_idx, int* __restrict__ knn_out) {
  const int qt  = blockIdx.x;   // query tile 0..63
  const int b   = blockIdx.y;   // cloud 0..15
  const int tid = threadIdx.x;
  const int wave = tid >> 5;
  const int lane = tid & 31;
  const int mn   = lane & 15;
  const int koff = (lane >> 4) * 2;
  const int radd = (lane >> 4) * 8;

  __shared__ __align__(16) float sQ[16 * FIN];       // 4 KB
  __shared__ __align__(16) float sX[2][128 * FIN];   // 2 x 32 KB, double buffer
  __shared__ float sXn[128];
  __shared__ float sD[16 * 128];                     // 8 KB

  const float* xbase = feat + (size_t)b * Pc * FIN;

  // Prologue: start chunk 0 DMA, load Q tile synchronously meanwhile.
  copy_chunk(xbase, sX[0], tid);
  for (int i = tid; i < 16 * FIN; i += 256) {
    int r = i >> 6, k = i & 63;
    int ql = fps_idx[b * Mc + qt * 16 + r];
    sQ[i] = feat[(size_t)(b * Pc + ql) * FIN + k];
  }
  async_wait();
  __syncthreads();

  float bestd[Kc];
  int   besti[Kc];
#pragma unroll
  for (int i = 0; i < Kc; ++i) { bestd[i] = FMAXV; besti[i] = 0; }

  for (int c = 0; c < Pc / 128; ++c) {
    const int buf = c & 1;
    // Kick off next chunk DMA; it overlaps WMMA + selection below.
    if (c + 1 < Pc / 128)
      copy_chunk(xbase + (size_t)(c + 1) * 128 * FIN, sX[buf ^ 1], tid);
    if (tid < 128) sXn[tid] = xnorm[(size_t)b * Pc + c * 128 + tid];
    __syncthreads();

    const float* xb = sX[buf];
    v8f acc = {};
    for (int k0 = 0; k0 < FIN; k0 += 4) {
      v2f a, bf;
      a.x  = sQ[mn * FIN + k0 + koff];
      a.y  = sQ[mn * FIN + k0 + koff + 1];
      bf.x = xb[(wave * 16 + mn) * FIN + k0 + koff];
      bf.y = xb[(wave * 16 + mn) * FIN + k0 + koff + 1];
      acc = __builtin_amdgcn_wmma_f32_16x16x4_f32(false, a, false, bf,
                                                  (short)0, acc, false, false);
    }
    const float xn = sXn[wave * 16 + mn];
#pragma unroll
    for (int i = 0; i < 8; ++i) {
      sD[(i + radd) * 128 + wave * 16 + mn] = xn - 2.0f * acc[i];
    }
    __syncthreads();

    if (tid < 16) {  // one thread per query maintains its sorted top-K list
      for (int p = 0; p < 128; ++p) {
        float d = sD[tid * 128 + p];
        if (d < bestd[Kc - 1]) {
          int j = Kc - 1;
          while (j > 0 && bestd[j - 1] > d) {
            bestd[j] = bestd[j - 1]; besti[j] = besti[j - 1]; --j;
          }
          bestd[j] = d; besti[j] = c * 128 + p;
        }
      }
    }
    async_wait();     // next buffer resident before we flip
    __syncthreads();  // also: sD consumed before next WMMA overwrites it
  }

  if (tid < 16) {
    int qrow = b * Mc + qt * 16 + tid;
#pragma unroll
    for (int k = 0; k < Kc; ++k)
      knn_out[(size_t)qrow * Kc + k] = b * Pc + besti[k];
  }
}

// ---------------------------------------------------------------------------
// 4) Gather neighbor features (post-BN/ReLU) and max-pool over K.
// ---------------------------------------------------------------------------
__global__ __launch_bounds__(128)
void gather_max_kernel(const float* __restrict__ h, const int* __restrict__ knn,
                       float* __restrict__ out_feat) {
  const int row = blockIdx.x;
  const int c   = threadIdx.x;
  int gidx[Kc];
#pragma unroll
  for (int k = 0; k < Kc; ++k) {
    gidx[k] = knn[row * Kc + k];
    __builtin_prefetch(&h[(size_t)gidx[k] * FOUT + c], 0, 0);
  }
  float m = -FMAXV;
#pragma unroll
  for (int k = 0; k < Kc; ++k)
    m = fmaxf(m, h[(size_t)gidx[k] * FOUT + c]);
  out_feat[(size_t)row * FOUT + c] = m;
}

// ---------------------------------------------------------------------------
extern "C" void kernel_launch(void* const* d_in, const int* in_sizes, int n_in,
                              void* d_out, int out_size, void* d_ws, size_t ws_size,
                              hipStream_t stream) {
  (void)in_sizes; (void)n_in; (void)out_size; (void)ws_size;
  const float* pos   = (const float*)d_in[0];
  const float* feat  = (const float*)d_in[1];
  /* d_in[2] = batch_indices (layout known: sorted, equal clouds) */
  const float* W     = (const float*)d_in[3];
  const float* bias  = (const float*)d_in[4];
  const float* gamma = (const float*)d_in[5];
  const float* beta  = (const float*)d_in[6];

  char* ws = (char*)d_ws;
  int*   fpsIdx = (int*)ws;     ws += (size_t)Bc * Mc * sizeof(int);
  float* h      = (float*)ws;   ws += (size_t)Nc * FOUT * sizeof(float);
  float* g_sum  = (float*)ws;   ws += FOUT * sizeof(float);
  float* g_sq   = (float*)ws;   ws += FOUT * sizeof(float);
  float* scale  = (float*)ws;   ws += FOUT * sizeof(float);
  float* shift  = (float*)ws;   ws += FOUT * sizeof(float);
  float* xnorm  = (float*)ws;   ws += (size_t)Nc * sizeof(float);
  int*   knn    = (int*)ws;     ws += (size_t)Bc * Mc * Kc * sizeof(int);

  float* out_feat  = (float*)d_out;
  float* out_pos   = out_feat + (size_t)Bc * Mc * FOUT;
  int*   out_batch = (int*)(out_pos + (size_t)Bc * Mc * 3);

  hipLaunchKernelGGL(fps_kernel, dim3(Bc), dim3(1024), 0, stream,
                     pos, fpsIdx, out_pos, out_batch);
  hipLaunchKernelGGL(init_bn_kernel, dim3(1), dim3(FOUT), 0, stream, g_sum, g_sq);
  hipLaunchKernelGGL(gemm_bn_kernel, dim3(256), dim3(256), 0, stream,
                     feat, W, bias, h, g_sum, g_sq);
  hipLaunchKernelGGL(bn_finalize_kernel, dim3(1), dim3(FOUT), 0, stream,
                     g_sum, g_sq, gamma, beta, scale, shift);
  hipLaunchKernelGGL(bn_apply_kernel, dim3(((size_t)Nc * FOUT) / (256 * 4)),
                     dim3(256), 0, stream, h, scale, shift);
  hipLaunchKernelGGL(xnorm_kernel, dim3(Nc / 256), dim3(256), 0, stream,
                     feat, xnorm);
  hipLaunchKernelGGL(knn_kernel, dim3(Mc / 16, Bc), dim3(256), 0, stream,
                     feat, xnorm, fpsIdx, knn);
  hipLaunchKernelGGL(gather_max_kernel, dim3(Bc * Mc), dim3(FOUT), 0, stream,
                     h, knn, out_feat);
}